// get_model_16922171146624
// MI455X (gfx1250) — compile-verified
//
#include <hip/hip_runtime.h>

typedef __attribute__((ext_vector_type(2))) float v2f;
typedef __attribute__((ext_vector_type(8))) float v8f;

#define NB   8
#define NP   2048
#define KNN  20
#define MTOT (NB*NP*KNN)   // 327680, divisible by 256 and 16

__device__ __forceinline__ float leaky(float v) { return v >= 0.f ? v : 0.2f * v; }

// ---------------------------------------------------------------------------
// Kernel 1: KNN (top-20 of pd = 2*x_i.x_j - |x_i|^2 - |x_j|^2, includes self).
// One block of 256 threads per 256 rows of one batch; x[b] staged in LDS.
// Per-thread top-20 kept in registers via fully-unrolled compare-swap chain
// (stable w.r.t. ties, matching jax top_k index ordering).
// ---------------------------------------------------------------------------
__global__ void __launch_bounds__(256) knn_kernel(const float* __restrict__ x,
                                                  int* __restrict__ idx) {
  __shared__ float xs[3 * NP];
  __shared__ float xx[NP];
  const int b  = blockIdx.x >> 3;
  const int n0 = (blockIdx.x & 7) * 256;
  for (int t = threadIdx.x; t < 3 * NP; t += 256) xs[t] = x[(size_t)b * 3 * NP + t];
  __syncthreads();
  for (int j = threadIdx.x; j < NP; j += 256) {
    float a0 = xs[j], a1 = xs[NP + j], a2 = xs[2 * NP + j];
    xx[j] = a0 * a0 + a1 * a1 + a2 * a2;
  }
  __syncthreads();

  const int   n   = n0 + threadIdx.x;
  const float xi0 = xs[n], xi1 = xs[NP + n], xi2 = xs[2 * NP + n];
  const float xxi = xx[n];

  float bd[KNN]; int bi[KNN];
#pragma unroll
  for (int t = 0; t < KNN; ++t) { bd[t] = -3.4e38f; bi[t] = 0; }

  for (int j = 0; j < NP; ++j) {
    float d = 2.f * (xi0 * xs[j] + xi1 * xs[NP + j] + xi2 * xs[2 * NP + j]) - xxi - xx[j];
    if (d > bd[KNN - 1]) {
      float cd = d; int cj = j;
#pragma unroll
      for (int t = 0; t < KNN; ++t) {
        if (cd > bd[t]) {
          float td = bd[t]; bd[t] = cd; cd = td;
          int   ti = bi[t]; bi[t] = cj; cj = ti;
        }
      }
    }
  }
  int* op = idx + (size_t)(b * NP + n) * KNN;
#pragma unroll
  for (int t = 0; t < KNN; ++t) op[t] = bi[t];
}

// ---------------------------------------------------------------------------
// Kernel 2: edge features e[6] = {nbr - ctr, ctr}, then MLP0 (6->64) + BN + leaky.
// One thread per edge m = ((b*N + n)*K + k). Writes E (M x 6) and Y (M x 64).
// ---------------------------------------------------------------------------
__global__ void __launch_bounds__(256) edge_mlp0_kernel(
    const float* __restrict__ x, const int* __restrict__ idx,
    const float* __restrict__ W0, const float* __restrict__ s0,
    const float* __restrict__ b0, float* __restrict__ E, float* __restrict__ Y) {
  const int m = blockIdx.x * 256 + threadIdx.x;
  if (m >= MTOT) return;
  const int n = (m / KNN) % NP;
  const int b = m / (KNN * NP);
  const int j = idx[m];

  float e[6];
#pragma unroll
  for (int c = 0; c < 3; ++c) {
    float cx = x[((size_t)b * 3 + c) * NP + n];
    float nx = x[((size_t)b * 3 + c) * NP + j];
    e[c]     = nx - cx;
    e[3 + c] = cx;
  }
  float* ep = E + (size_t)m * 6;
#pragma unroll
  for (int i = 0; i < 6; ++i) ep[i] = e[i];

  float* yp = Y + (size_t)m * 64;
  for (int o = 0; o < 64; ++o) {
    float acc = 0.f;
#pragma unroll
    for (int i = 0; i < 6; ++i) acc += e[i] * W0[o * 6 + i];
    yp[o] = leaky(acc * s0[o] + b0[o]);
  }
}

// ---------------------------------------------------------------------------
// Kernel 3: heavy GEMM + e-contraction + BN1 + leaky via fp32 WMMA.
// Reformulated as ONE GEMM with K = 6*64 = 384:
//   Ahat[m, 64*i + k] = e[m,i] * Y[m,k]      (e-scale folded into A)
//   Bhat[o, 64*i + k] = W1[6*o + i, k]
//   out[m,o] = leaky(s1[o] * (Ahat @ Bhat^T)[m,o] + b1[o])
// The A-fragment row is lane%16 for ALL lanes (16x4 f32 A layout), so e[m,i]
// is a per-lane scalar: scaling A costs 6*32 VALU mults per lane, and the
// whole reduction accumulates directly through the WMMA C/D operand.
// One wave per 16-row M-tile; 4 independent accumulator chains (one per
// 16-col N-tile) interleaved across the 96 k-steps. EXEC all-ones (exact
// grid, no divergence) as WMMA requires.
// ---------------------------------------------------------------------------
__global__ void __launch_bounds__(256) gemm_wmma_kernel(
    const float* __restrict__ Y,  const float* __restrict__ E,
    const float* __restrict__ W1, const float* __restrict__ s1,
    const float* __restrict__ b1, float* __restrict__ OUT1) {
  const int wave = threadIdx.x >> 5;
  const int lane = threadIdx.x & 31;
  const int row  = lane & 15;   // A row (all lanes) / D col within tile
  const int half = lane >> 4;   // K sub-offset select per ISA layout
  const int tile = blockIdx.x * 8 + wave;
  const int m0   = tile * 16;

  // A fragments: a[ks] holds Y[m0+row][4*ks + 2*half + {0,1}]
  v2f a[16];
  const float* yrow = Y + (size_t)(m0 + row) * 64 + 2 * half;
#pragma unroll
  for (int ks = 0; ks < 16; ++ks) a[ks] = *(const v2f*)(yrow + 4 * ks);

  // Per-lane e scalars for this lane's A row.
  float es[6];
  {
    const float* erow = E + (size_t)(m0 + row) * 6;
    v2f t0 = *(const v2f*)(erow);
    v2f t1 = *(const v2f*)(erow + 2);
    v2f t2 = *(const v2f*)(erow + 4);
    es[0] = t0[0]; es[1] = t0[1]; es[2] = t1[0];
    es[3] = t1[1]; es[4] = t2[0]; es[5] = t2[1];
  }

  v8f oac[4];
#pragma unroll
  for (int nt = 0; nt < 4; ++nt)
    oac[nt] = (v8f){0.f, 0.f, 0.f, 0.f, 0.f, 0.f, 0.f, 0.f};

#pragma unroll
  for (int i = 0; i < 6; ++i) {
    // e-scaled A fragments for this i (reused across all 4 N-tiles)
    v2f ai[16];
#pragma unroll
    for (int ks = 0; ks < 16; ++ks) {
      v2f t; t[0] = a[ks][0] * es[i]; t[1] = a[ks][1] * es[i];
      ai[ks] = t;
    }
#pragma unroll
    for (int nt = 0; nt < 4; ++nt) {
      // B col n = 16*nt + row  ->  W1 row 6*n + i ; K layout mirrors A
      const float* brow = W1 + (size_t)(6 * (16 * nt + row) + i) * 64 + 2 * half;
#pragma unroll
      for (int ks = 0; ks < 16; ++ks) {
        v2f bf = *(const v2f*)(brow + 4 * ks);
        oac[nt] = __builtin_amdgcn_wmma_f32_16x16x4_f32(
            /*neg_a=*/false, ai[ks], /*neg_b=*/false, bf,
            /*c_mod=*/(short)0, oac[nt], /*reuse_a=*/false, /*reuse_b=*/false);
      }
    }
  }

  // Fused BN1 + leaky epilogue; column o = 16*nt + row is lane-constant.
  // D layout: VGPR v -> row (m0 + v + 8*half), col = 16*nt + row.
#pragma unroll
  for (int nt = 0; nt < 4; ++nt) {
    const int   o  = 16 * nt + row;
    const float sc = s1[o], bs = b1[o];
#pragma unroll
    for (int v = 0; v < 8; ++v) {
      float t = leaky(oac[nt][v] * sc + bs);
      OUT1[(size_t)(m0 + v + 8 * half) * 64 + o] = t;
    }
  }
}

// ---------------------------------------------------------------------------
// Kernel 4: max over K=20 neighbors. One thread per (b,n,o); coalesced reads.
// ---------------------------------------------------------------------------
__global__ void __launch_bounds__(256) maxk_kernel(const float* __restrict__ OUT1,
                                                   float* __restrict__ X1) {
  const int t  = blockIdx.x * 256 + threadIdx.x;  // < NB*NP*64
  const int o  = t & 63;
  const int bn = t >> 6;
  const float* p = OUT1 + (size_t)bn * KNN * 64 + o;
  float m = p[0];
#pragma unroll
  for (int k = 1; k < KNN; ++k) m = fmaxf(m, p[(size_t)k * 64]);
  X1[t] = m;
}

// ---------------------------------------------------------------------------
// Kernel 5: 64 -> 3 head + BN + leaky, output transposed to (B, 3, N).
// ---------------------------------------------------------------------------
__global__ void __launch_bounds__(256) final_kernel(
    const float* __restrict__ X1, const float* __restrict__ Wc,
    const float* __restrict__ sc, const float* __restrict__ bc,
    float* __restrict__ out) {
  const int bn = blockIdx.x * 256 + threadIdx.x;  // < NB*NP
  const int b = bn / NP, n = bn % NP;
  const float* xp = X1 + (size_t)bn * 64;
#pragma unroll
  for (int c = 0; c < 3; ++c) {
    float acc = 0.f;
    for (int o = 0; o < 64; ++o) acc += xp[o] * Wc[c * 64 + o];
    out[((size_t)b * 3 + c) * NP + n] = leaky(acc * sc[c] + bc[c]);
  }
}

// ---------------------------------------------------------------------------
extern "C" void kernel_launch(void* const* d_in, const int* in_sizes, int n_in,
                              void* d_out, int out_size, void* d_ws, size_t ws_size,
                              hipStream_t stream) {
  const float* x     = (const float*)d_in[0];
  const float* W0    = (const float*)d_in[1];
  const float* bn0_s = (const float*)d_in[2];
  const float* bn0_b = (const float*)d_in[3];
  const float* W1    = (const float*)d_in[4];
  const float* bn1_s = (const float*)d_in[5];
  const float* bn1_b = (const float*)d_in[6];
  const float* Wc    = (const float*)d_in[7];
  const float* bnc_s = (const float*)d_in[8];
  const float* bnc_b = (const float*)d_in[9];
  float* out = (float*)d_out;

  // Workspace carve-up (all offsets 256B-aligned):
  //   idx : MTOT int32            =   1,310,720 B   @ 0
  //   E   : MTOT x 6  f32         =   7,864,320 B   @ 1,310,720
  //   Y   : MTOT x 64 f32         =  83,886,080 B   @ 9,175,040
  //   O1  : MTOT x 64 f32         =  83,886,080 B   @ 93,061,120
  //   X1  : NB*NP x 64 f32        =   4,194,304 B   @ 176,947,200
  char*  ws  = (char*)d_ws;
  int*   idx = (int*)(ws);
  float* E   = (float*)(ws + 1310720);
  float* Y   = (float*)(ws + 9175040);
  float* O1  = (float*)(ws + 93061120);
  float* X1  = (float*)(ws + 176947200);

  knn_kernel      <<<NB * (NP / 256), 256, 0, stream>>>(x, idx);
  edge_mlp0_kernel<<<MTOT / 256,      256, 0, stream>>>(x, idx, W0, bn0_s, bn0_b, E, Y);
  gemm_wmma_kernel<<<MTOT / (16 * 8), 256, 0, stream>>>(Y, E, W1, bn1_s, bn1_b, O1);
  maxk_kernel     <<<(NB * NP * 64) / 256, 256, 0, stream>>>(O1, X1);
  final_kernel    <<<(NB * NP) / 256, 256, 0, stream>>>(X1, Wc, bnc_s, bnc_b, out);
}